// Net_29214367548090
// MI455X (gfx1250) — compile-verified
//
#include <hip/hip_runtime.h>
#include <hip/hip_bf16.h>
#include <math.h>

// ---------------------------------------------------------------------------
// SplineConv (K=2, linear basis) x2 + ELU + log_softmax for MI455X (gfx1250).
//
// Strategy: factor the per-edge basis mix out of the GEMM. Precompute per-node
//   y0 = X @ W[0], y1 = X @ W[1], r = X @ root + b      (WMMA f32 16x16x4)
// then the per-edge work is a 32-float gather + blend + 16 float atomics.
// Working set (~60MB) is L2-resident on MI455X (192MB L2), so the scatter
// phase runs at L2/atomic speed, not HBM speed.
//
// N_NODES == 100000 == 16 * 6250: every 16-row WMMA tile is full, so the
// node-transform inner loop needs no predication (straight-line clause loads).
// ---------------------------------------------------------------------------

#define N_NODES 100000
#define N_EDGES 1600000

typedef __attribute__((ext_vector_type(2))) float v2f;
typedef __attribute__((ext_vector_type(8))) float v8f;

// ---------------------------------------------------------------------------
// Per-node transform: one wave owns a 16-row tile of the input matrix and
// produces three 16-wide outputs with V_WMMA_F32_16X16X4_F32 (fp32-exact).
//   ybuf[n][0..15]  = (X @ Wa)[n]        (basis k=0 features)
//   ybuf[n][16..31] = (X @ Wb)[n]        (basis k=1 features)
//   rbuf[n][0..15]  = (X @ Wr)[n] + bias (root/residual term)
//
// A-fragment (16x4 f32):  lane(0..15)=M, comp0=K0/K2(hi), comp1=K1/K3(hi)
// B-fragment (4x16 f32):  lane&15=N,     comp0=K0/K2(hi), comp1=K1/K3(hi)
// D (16x16 f32): comp v -> M = v + 8*hi, N = lane&15
// ---------------------------------------------------------------------------
template <int KDIM>
__global__ void node_transform_wmma(const float* __restrict__ X,
                                    const float* __restrict__ Wa,
                                    const float* __restrict__ Wb,
                                    const float* __restrict__ Wr,
                                    const float* __restrict__ bias,
                                    float* __restrict__ ybuf,   // [N][32]
                                    float* __restrict__ rbuf,   // [N][16]
                                    int n_nodes) {
  const int wave = (int)((blockIdx.x * blockDim.x + threadIdx.x) >> 5);
  const int lane = (int)(threadIdx.x & 31);
  const int row0 = wave * 16;
  if (row0 >= n_nodes) return;      // N_NODES % 16 == 0: live waves are full

  const int m  = lane & 15;   // N column (B/C/D), M row (A)
  const int hi = lane >> 4;   // half-wave selects K pair

  v8f acc0 = {};  // X @ Wa
  v8f acc1 = {};  // X @ Wb
  v8f accr = {};  // X @ Wr

  const float* xrow = X + (size_t)(row0 + m) * KDIM;

#pragma unroll
  for (int s = 0; s < KDIM / 4; ++s) {
    const int k = s * 4 + hi * 2;   // this lane's K pair (always even -> 8B aligned)

    const v2f a = *(const v2f*)(xrow + k);   // global_load_b64

    v2f b0, b1, br;                 // weights are tiny -> L0/L2 resident
    b0.x = Wa[k * 16 + m];  b0.y = Wa[(k + 1) * 16 + m];
    b1.x = Wb[k * 16 + m];  b1.y = Wb[(k + 1) * 16 + m];
    br.x = Wr[k * 16 + m];  br.y = Wr[(k + 1) * 16 + m];

    acc0 = __builtin_amdgcn_wmma_f32_16x16x4_f32(false, a, false, b0,
                                                 (short)0, acc0, false, false);
    acc1 = __builtin_amdgcn_wmma_f32_16x16x4_f32(false, a, false, b1,
                                                 (short)0, acc1, false, false);
    accr = __builtin_amdgcn_wmma_f32_16x16x4_f32(false, a, false, br,
                                                 (short)0, accr, false, false);
  }

  const float bv = bias[m];
#pragma unroll
  for (int v = 0; v < 8; ++v) {
    const int r = row0 + v + hi * 8;
    ybuf[(size_t)r * 32 + m]      = acc0[v];
    ybuf[(size_t)r * 32 + 16 + m] = acc1[v];
    rbuf[(size_t)r * 16 + m]      = accr[v] + bv;
  }
}

// ---------------------------------------------------------------------------
// Edge scatter: 4 threads per edge, each thread handles 4 of the 16 output
// channels. msg = (1-u)*y0[src] + u*y1[src]; atomicAdd into agg[dst].
// agg atomics land in L2 (entire agg buffer is 6.4MB, L2-resident).
// ---------------------------------------------------------------------------
__global__ void edge_scatter(const float* __restrict__ ybuf,     // [N][32]
                             const long long* __restrict__ ei,   // [2][E] i64
                             const float* __restrict__ ea,       // [E]
                             float* __restrict__ agg,            // [N][16]
                             float* __restrict__ cnt,            // [N]
                             int n_edges, int add_cnt) {
  const long long gid = (long long)blockIdx.x * blockDim.x + threadIdx.x;
  const long long e   = gid >> 2;
  const int part      = (int)(gid & 3);
  if (e >= n_edges) return;

  const long long src = ei[e];
  const long long dst = ei[(long long)n_edges + e];
  const float u  = ea[e];
  const float w0 = 1.0f - u;

  const float4* yrow = (const float4*)(ybuf + src * 32);
  const float4 a = yrow[part];        // y0 chunk  (global_load_b128)
  const float4 b = yrow[part + 4];    // y1 chunk

  float4 msg;
  msg.x = w0 * a.x + u * b.x;
  msg.y = w0 * a.y + u * b.y;
  msg.z = w0 * a.z + u * b.z;
  msg.w = w0 * a.w + u * b.w;

  float* ag = agg + dst * 16 + part * 4;
  atomicAdd(&ag[0], msg.x);
  atomicAdd(&ag[1], msg.y);
  atomicAdd(&ag[2], msg.z);
  atomicAdd(&ag[3], msg.w);

  if (add_cnt && part == 0) atomicAdd(&cnt[dst], 1.0f);
}

// ---------------------------------------------------------------------------
// Layer-1 finish: h = elu(agg/max(cnt,1) + r). 4 threads per node.
// ---------------------------------------------------------------------------
__device__ __forceinline__ float elu1(float x) {
  return x > 0.0f ? x : (__expf(x) - 1.0f);
}

__global__ void node_elu(const float* __restrict__ agg,
                         const float* __restrict__ cnt,
                         const float* __restrict__ rbuf,
                         float* __restrict__ h, int n_nodes) {
  const int gid  = (int)(blockIdx.x * blockDim.x + threadIdx.x);
  const int node = gid >> 2;
  const int part = gid & 3;
  if (node >= n_nodes) return;

  const float inv = 1.0f / fmaxf(cnt[node], 1.0f);
  const float4 a = ((const float4*)(agg + (size_t)node * 16))[part];
  const float4 r = ((const float4*)(rbuf + (size_t)node * 16))[part];
  float4 o;
  o.x = elu1(a.x * inv + r.x);
  o.y = elu1(a.y * inv + r.y);
  o.z = elu1(a.z * inv + r.z);
  o.w = elu1(a.w * inv + r.w);
  ((float4*)(h + (size_t)node * 16))[part] = o;
}

// ---------------------------------------------------------------------------
// Layer-2 finish: out = log_softmax(agg/max(cnt,1) + r, axis=1). One thread
// per node (16 channels).
// ---------------------------------------------------------------------------
__global__ void node_final(const float* __restrict__ agg,
                           const float* __restrict__ cnt,
                           const float* __restrict__ rbuf,
                           float* __restrict__ out, int n_nodes) {
  const int node = (int)(blockIdx.x * blockDim.x + threadIdx.x);
  if (node >= n_nodes) return;

  const float inv = 1.0f / fmaxf(cnt[node], 1.0f);
  float v[16];
  float mx = -INFINITY;
#pragma unroll
  for (int c = 0; c < 16; ++c) {
    const float t = agg[(size_t)node * 16 + c] * inv + rbuf[(size_t)node * 16 + c];
    v[c] = t;
    mx   = fmaxf(mx, t);
  }
  float s = 0.0f;
#pragma unroll
  for (int c = 0; c < 16; ++c) s += __expf(v[c] - mx);
  const float lse = mx + __logf(s);
#pragma unroll
  for (int c = 0; c < 16; ++c) out[(size_t)node * 16 + c] = v[c] - lse;
}

// ---------------------------------------------------------------------------
// Host-side orchestration (graph-capture safe: only async ops on `stream`).
// ---------------------------------------------------------------------------
extern "C" void kernel_launch(void* const* d_in, const int* in_sizes, int n_in,
                              void* d_out, int out_size, void* d_ws, size_t ws_size,
                              hipStream_t stream) {
  (void)in_sizes; (void)n_in; (void)out_size; (void)ws_size;

  const float*     x     = (const float*)d_in[0];          // [100000,48]
  const float*     ea    = (const float*)d_in[1];          // [1600000,1]
  const long long* ei    = (const long long*)d_in[2];      // [2,1600000] i64
  const float*     W1    = (const float*)d_in[3];          // [2,48,16]
  const float*     root1 = (const float*)d_in[4];          // [48,16]
  const float*     b1    = (const float*)d_in[5];          // [16]
  const float*     W2    = (const float*)d_in[6];          // [2,16,16]
  const float*     root2 = (const float*)d_in[7];          // [16,16]
  const float*     b2    = (const float*)d_in[8];          // [16]
  float*           out   = (float*)d_out;                  // [100000,16]

  // Workspace layout (floats): ybuf | rbuf | agg | cnt | h   (~32.4 MB)
  float* ws   = (float*)d_ws;
  float* ybuf = ws;                                  // 100000*32
  float* rbuf = ybuf + (size_t)N_NODES * 32;         // 100000*16
  float* agg  = rbuf + (size_t)N_NODES * 16;         // 100000*16
  float* cnt  = agg  + (size_t)N_NODES * 16;         // 100000
  float* h    = cnt  + (size_t)N_NODES;              // 100000*16

  const int waves_needed = (N_NODES + 15) / 16;                  // 6250 waves
  const int nt_blocks    = (waves_needed + 7) / 8;               // 8 waves/block
  const int es_threads   = N_EDGES * 4;
  const int es_blocks    = (es_threads + 255) / 256;
  const int elu_blocks   = (N_NODES * 4 + 255) / 256;
  const int fin_blocks   = (N_NODES + 255) / 256;

  // ---- Layer 1 ----
  hipMemsetAsync(agg, 0, ((size_t)N_NODES * 16 + N_NODES) * sizeof(float), stream);

  node_transform_wmma<48><<<nt_blocks, 256, 0, stream>>>(
      x, W1, W1 + 48 * 16, root1, b1, ybuf, rbuf, N_NODES);

  edge_scatter<<<es_blocks, 256, 0, stream>>>(
      ybuf, ei, ea, agg, cnt, N_EDGES, /*add_cnt=*/1);

  node_elu<<<elu_blocks, 256, 0, stream>>>(agg, cnt, rbuf, h, N_NODES);

  // ---- Layer 2 (reuse ybuf/rbuf/agg; cnt is identical for both layers) ----
  hipMemsetAsync(agg, 0, (size_t)N_NODES * 16 * sizeof(float), stream);

  node_transform_wmma<16><<<nt_blocks, 256, 0, stream>>>(
      h, W2, W2 + 16 * 16, root2, b2, ybuf, rbuf, N_NODES);

  edge_scatter<<<es_blocks, 256, 0, stream>>>(
      ybuf, ei, ea, agg, cnt, N_EDGES, /*add_cnt=*/0);

  node_final<<<fin_blocks, 256, 0, stream>>>(agg, cnt, rbuf, out, N_NODES);
}